// Determ_30846455120059
// MI455X (gfx1250) — compile-verified
//
#include <hip/hip_runtime.h>

// ---------------------------------------------------------------------------
// Jacobian-determinant penalty for a [2,3,160,192,224] f32 displacement field.
// Memory-bound: 165 MB input (fits in 192 MB L2), ~7 us HBM floor @ 23.3 TB/s.
// Compute is negligible (~0.6 GFLOP). WMMA (f32 16x16x4) is used only for the
// exact wave-level sum reduction in the epilogue to exercise the matrix pipe.
// ---------------------------------------------------------------------------

typedef __attribute__((ext_vector_type(2))) float v2f;
typedef __attribute__((ext_vector_type(8))) float v8f;

#define BB   2
#define CD   160
#define CH   192
#define CW   224
#define CHW  (CH * CW)        // 43008
#define CDHW (CD * CHW)       // 6881280
#define NVOX (BB * CDHW)      // 13762560 voxels averaged over
#define NBLK 2048
#define NTHR 256

// Exact f32 wave(32)-sum using V_WMMA_F32_16X16X4_F32.
// A = all-ones (16x4); each lane contributes its value exactly once into B.
// Sum over ALL 256 elements of D = sum_{m,n,k} A[m][k]*B[k][n] = 16 * sum(B),
// which is independent of the precise VGPR striping of A/B/D. Each lane sums
// its 8 D registers, a shfl_xor tree sums across lanes, then /16 (exact).
__device__ __forceinline__ float wave_sum_wmma(float v) {
  v2f a; a[0] = 1.0f; a[1] = 1.0f;   // ones matrix
  v2f b; b[0] = v;    b[1] = 0.0f;   // one contribution per lane
  v8f c = {};
  v8f d = __builtin_amdgcn_wmma_f32_16x16x4_f32(
      /*neg_a=*/false, a, /*neg_b=*/false, b,
      /*c_mod=*/(short)0, c, /*reuse_a=*/false, /*reuse_b=*/false);
  float s = d[0] + d[1] + d[2] + d[3] + d[4] + d[5] + d[6] + d[7];
  s += __shfl_xor(s, 16);
  s += __shfl_xor(s, 8);
  s += __shfl_xor(s, 4);
  s += __shfl_xor(s, 2);
  s += __shfl_xor(s, 1);
  return s * (1.0f / 16.0f);         // exact power-of-two scale
}

__global__ __launch_bounds__(NTHR) void jdet_partial(const float* __restrict__ p,
                                                     float* __restrict__ partials) {
  float acc = 0.0f;
  const int stride = NBLK * NTHR;
  for (int idx = (int)blockIdx.x * NTHR + (int)threadIdx.x; idx < NVOX; idx += stride) {
    // Decompose flat index -> (b,d,h,w); lanes are consecutive in w (coalesced).
    int w = idx % CW;
    int t = idx / CW;
    int h = t % CH;
    t /= CH;
    int d = t % CD;
    int b = t / CD;

    // np.gradient edge handling: clamped taps + scale.
    // interior: (f[i+1]-f[i-1])*0.5 ; edges: one-sided difference.
    int dm = d > 0 ? d - 1 : 0, dp = d < CD - 1 ? d + 1 : CD - 1;
    int hm = h > 0 ? h - 1 : 0, hp = h < CH - 1 ? h + 1 : CH - 1;
    int wm = w > 0 ? w - 1 : 0, wp = w < CW - 1 ? w + 1 : CW - 1;
    float sd = (dp - dm == 2) ? 0.5f : 1.0f;
    float sh = (hp - hm == 2) ? 0.5f : 1.0f;
    float sw = (wp - wm == 2) ? 0.5f : 1.0f;

    float gd[3], gh[3], gw[3];
    const float* q = p + (long)b * (3 * CDHW);
    const int ohw = h * CW + w;
    const int od  = d * CHW;
#pragma unroll
    for (int cch = 0; cch < 3; ++cch) {
      const float* qc = q + cch * CDHW;
      gd[cch] = (qc[dp * CHW + ohw]   - qc[dm * CHW + ohw])   * sd;
      gh[cch] = (qc[od + hp * CW + w] - qc[od + hm * CW + w]) * sh;
      gw[cch] = (qc[od + h * CW + wp] - qc[od + h * CW + wm]) * sw;
    }
    // identity grid contribution: d(grid)/daxis == 1 on the diagonal channel
    gd[0] += 1.0f;
    gh[1] += 1.0f;
    gw[2] += 1.0f;

    // 3x3 determinant, cofactor expansion along the first row (matches ref).
    float det = gd[0] * (gh[1] * gw[2] - gh[2] * gw[1])
              - gd[1] * (gh[0] * gw[2] - gh[2] * gw[0])
              + gd[2] * (gh[0] * gw[1] - gh[1] * gw[0]);
    float m = det < 0.0f ? det : 0.0f;   // clamp(max=0)
    acc = fmaf(m, m, acc);
  }

  // All lanes converged here (full blocks, grid-stride loop): EXEC is all 1s.
  float wsum = wave_sum_wmma(acc);

  __shared__ float ws[NTHR / 32];
  const int lane = threadIdx.x & 31;
  const int wid  = threadIdx.x >> 5;
  if (lane == 0) ws[wid] = wsum;
  __syncthreads();
  if (threadIdx.x == 0) {
    float s = 0.0f;
#pragma unroll
    for (int i = 0; i < NTHR / 32; ++i) s += ws[i];
    partials[blockIdx.x] = s;            // fully overwritten every call
  }
}

__global__ __launch_bounds__(NTHR) void jdet_final(const float* __restrict__ partials,
                                                   float* __restrict__ out) {
  float s = 0.0f;
  for (int i = threadIdx.x; i < NBLK; i += NTHR) s += partials[i];

  float wsum = wave_sum_wmma(s);

  __shared__ float ws[NTHR / 32];
  const int lane = threadIdx.x & 31;
  const int wid  = threadIdx.x >> 5;
  if (lane == 0) ws[wid] = wsum;
  __syncthreads();
  if (threadIdx.x == 0) {
    float t = 0.0f;
#pragma unroll
    for (int i = 0; i < NTHR / 32; ++i) t += ws[i];
    out[0] = t / (float)NVOX;            // mean
  }
}

extern "C" void kernel_launch(void* const* d_in, const int* in_sizes, int n_in,
                              void* d_out, int out_size, void* d_ws, size_t ws_size,
                              hipStream_t stream) {
  (void)in_sizes; (void)n_in; (void)out_size; (void)ws_size;
  const float* y_pred = (const float*)d_in[0];
  float* out          = (float*)d_out;
  float* partials     = (float*)d_ws;    // NBLK * 4 bytes = 8 KB of scratch

  jdet_partial<<<NBLK, NTHR, 0, stream>>>(y_pred, partials);
  jdet_final<<<1, NTHR, 0, stream>>>(partials, out);
}